// ClementsPSBS_36094905156098
// MI455X (gfx1250) — compile-verified
//
#include <hip/hip_runtime.h>

// ---------------------------------------------------------------------------
// Clements mesh scan for MI455X (gfx1250, wave32).
//   - state (1024 complex features per batch row) lives in VGPRs:
//       1 wave  = 1 batch row, lane L owns features [32L, 32L+32)
//   - per-layer params (trig table + atten) double-buffered in LDS via
//     CDNA5 async global->LDS copies (s_wait_asynccnt), shared by 8 waves/WG
//   - trig (cos/sin theta, cos/sin(pi/4+split)) precomputed once into d_ws
// ---------------------------------------------------------------------------

#define NF 1024               // features
#define NL 1024               // layers
#define NA 512                // pairs per layer
#define WG_THREADS 256        // 8 waves
#define WAVES_PER_WG 8

// ---- CDNA5 async global->LDS path (probe-guarded, safe fallback) ----------
#if __has_builtin(__builtin_amdgcn_global_load_async_to_lds_b128)
#define HAS_ASYNC 1
#else
#define HAS_ASYNC 0
#endif

#if __has_builtin(__builtin_amdgcn_s_wait_asynccnt)
#define WAIT_ASYNC(n) __builtin_amdgcn_s_wait_asynccnt(n)
#elif HAS_ASYNC
#define WAIT_ASYNC(n) asm volatile("s_wait_asynccnt %0" ::"n"(n) : "memory")
#else
#define WAIT_ASYNC(n) do {} while (0)
#endif

typedef int v4i __attribute__((vector_size(16)));
typedef __attribute__((address_space(1))) v4i as1_v4i;  // global
typedef __attribute__((address_space(3))) v4i as3_v4i;  // LDS

__device__ __forceinline__ void async_cp16(const float* gsrc, float* ldst) {
#if HAS_ASYNC
  // AS1 pointer value == generic value; AS3 pointer == low 32 bits of generic
  // LDS aperture address, so integer round-trips give valid addrspace ptrs.
  __builtin_amdgcn_global_load_async_to_lds_b128(
      (as1_v4i*)(unsigned long long)gsrc,
      (as3_v4i*)(unsigned)(unsigned long long)ldst,
      0, 0);
#else
  *(float4*)ldst = *(const float4*)gsrc;
#endif
}

// ---- small complex helpers ------------------------------------------------
__device__ __forceinline__ void cmul(float2& u, float ar, float ai) {
  float r = u.x * ar - u.y * ai;
  float i = u.x * ai + u.y * ar;
  u.x = r; u.y = i;
}

// one MZI: pa = ua*e^{i th}; va = ca*pa + i sa*ub ; vb = i sa*pa + ca*ub
__device__ __forceinline__ void mzi(float2& ua, float2& ub,
                                    float ct, float st, float ca, float sa) {
  float par = ua.x * ct - ua.y * st;
  float pai = ua.x * st + ua.y * ct;
  float var = ca * par - sa * ub.y;
  float vai = ca * pai + sa * ub.x;
  float vbr = ca * ub.x - sa * pai;
  float vbi = ca * ub.y + sa * par;
  ua.x = var; ua.y = vai;
  ub.x = vbr; ub.y = vbi;
}

// ---- trig table prep: tab[l*2048 + p*4] = {cos th, sin th, cos a, sin a} --
__global__ __launch_bounds__(256) void trig_prep(const float* __restrict__ theta,
                                                 const float* __restrict__ split,
                                                 float* __restrict__ tab, int n) {
  int i = blockIdx.x * blockDim.x + threadIdx.x;
  if (i >= n) return;
  float th = theta[i];
  float sp = split[i];
  float st, ct;
  sincosf(th, &st, &ct);
  float sa, ca;
  sincosf(0.78539816339744830962f + sp, &sa, &ca);
  ((float4*)tab)[i] = make_float4(ct, st, ca, sa);
}

// ---- per-layer LDS staging ------------------------------------------------
// USE_TAB  layout: [0..2047] trig (512 pairs x float4), [2048..4095] atten
// !USE_TAB layout: [0..511] theta, [512..1023] split,   [1024..3071] atten
template <bool USE_TAB>
__device__ __forceinline__ void stage_layer(int l, float* buf,
                                            const float* trig_tab,
                                            const float* theta,
                                            const float* split,
                                            const float* atten) {
  constexpr int NCHUNK = USE_TAB ? 1024 : 768;  // float4 chunks
#pragma unroll
  for (int it = 0; it < NCHUNK / WG_THREADS; ++it) {
    int c = it * WG_THREADS + (int)threadIdx.x;
    const float* src;
    if (USE_TAB) {
      src = (c < 512) ? (trig_tab + (size_t)l * 2048 + c * 4)
                      : (atten + (size_t)l * 2048 + (c - 512) * 4);
    } else {
      if (c < 128)      src = theta + (size_t)l * 512 + c * 4;
      else if (c < 256) src = split + (size_t)l * 512 + (c - 128) * 4;
      else              src = atten + (size_t)l * 2048 + (c - 256) * 4;
    }
    async_cp16(src, buf + c * 4);
  }
}

template <bool USE_TAB>
__device__ __forceinline__ float4 get_trig(const float* tbuf, int lane, int j) {
  if (USE_TAB) {
    return ((const float4*)tbuf)[lane * 16 + j];
  } else {
    float th = tbuf[lane * 16 + j];
    float sp = tbuf[512 + lane * 16 + j];
    float st, ct;
    __sincosf(th, &st, &ct);
    float sa, ca;
    __sincosf(0.78539816339744830962f + sp, &sa, &ca);
    return make_float4(ct, st, ca, sa);
  }
}

// ---- main scan kernel -----------------------------------------------------
template <bool USE_TAB>
__global__ __launch_bounds__(WG_THREADS) void clements_scan(
    const float* __restrict__ x, const float* __restrict__ theta,
    const float* __restrict__ split, const float* __restrict__ atten,
    const float* __restrict__ trig_tab, float* __restrict__ out, int rows) {
  constexpr int BUF_FLOATS = USE_TAB ? 4096 : 3072;
  constexpr int CPL = USE_TAB ? 4 : 3;  // async copies per wave per layer
  __shared__ __align__(16) float smem[2][BUF_FLOATS];

  const int lane = threadIdx.x & 31;
  const int wave = threadIdx.x >> 5;
  const int row = blockIdx.x * WAVES_PER_WG + wave;
  const bool active = row < rows;

  // ---- load this row's state into registers (imag = 0) ----
  float2 s[32];
  if (active) {
    const float4* xr = (const float4*)(x + (size_t)row * NF + lane * 32);
#pragma unroll
    for (int k = 0; k < 8; ++k) {
      float4 v = xr[k];
      s[4 * k + 0] = make_float2(v.x, 0.f);
      s[4 * k + 1] = make_float2(v.y, 0.f);
      s[4 * k + 2] = make_float2(v.z, 0.f);
      s[4 * k + 3] = make_float2(v.w, 0.f);
    }
  } else {
#pragma unroll
    for (int k = 0; k < 32; ++k) s[k] = make_float2(0.f, 0.f);
  }

  // prologue: stage layer 0
  stage_layer<USE_TAB>(0, smem[0], trig_tab, theta, split, atten);

  for (int l = 0; l < NL; ++l) {
    float* buf = smem[l & 1];
    if (l + 1 < NL) {
      stage_layer<USE_TAB>(l + 1, smem[(l + 1) & 1], trig_tab, theta, split, atten);
      WAIT_ASYNC(CPL);  // waits for *this* layer's copies (issued last iter)
    } else {
      WAIT_ASYNC(0);
    }
    __syncthreads();  // layer-l LDS data visible to all waves

    const float* tbuf = buf;
    const float* abuf = buf + (USE_TAB ? 2048 : 1024);

    if ((l & 1) == 0) {
      // A layer: pairs (2i, 2i+1) -> fully intra-lane
#pragma unroll
      for (int j = 0; j < 16; ++j) {
        float4 t = get_trig<USE_TAB>(tbuf, lane, j);
        mzi(s[2 * j], s[2 * j + 1], t.x, t.y, t.z, t.w);
      }
    } else {
      // B layer: pairs (2i+1, 2i+2); 15 intra-lane + 1 cross-lane pair.
      // Features 0 and 1023 are untouched (wraparound pair is masked).
#pragma unroll
      for (int j = 0; j < 15; ++j) {
        float4 t = get_trig<USE_TAB>(tbuf, lane, j);
        mzi(s[2 * j + 1], s[2 * j + 2], t.x, t.y, t.z, t.w);
      }
      float4 t15 = get_trig<USE_TAB>(tbuf, lane, 15);
      // ub = neighbor lane's feature 0
      float nbr = __shfl_down(s[0].x, 1, 32);
      float nbi = __shfl_down(s[0].y, 1, 32);
      // pa from own feature 31
      float par = s[31].x * t15.x - s[31].y * t15.y;
      float pai = s[31].x * t15.y + s[31].y * t15.x;
      float var = t15.z * par - t15.w * nbi;
      float vai = t15.z * pai + t15.w * nbr;
      // vb for my feature 0 uses pa and (ca,sa) from lane-1's pair
      float paru = __shfl_up(par, 1, 32);
      float paiu = __shfl_up(pai, 1, 32);
      float cau = __shfl_up(t15.z, 1, 32);
      float sau = __shfl_up(t15.w, 1, 32);
      float vbr = cau * s[0].x - sau * paiu;
      float vbi = cau * s[0].y + sau * paru;
      if (lane < 31) { s[31].x = var; s[31].y = vai; }
      if (lane > 0)  { s[0].x = vbr; s[0].y = vbi; }
    }

    // per-feature complex attenuation (float4 = 2 complex values)
    {
      const float4* a4 = (const float4*)abuf + lane * 16;
#pragma unroll
      for (int k = 0; k < 16; ++k) {
        float4 av = a4[k];
        cmul(s[2 * k + 0], av.x, av.y);
        cmul(s[2 * k + 1], av.z, av.w);
      }
    }
    __syncthreads();  // all waves done reading buf before it is restaged
  }

  // ---- write complex64 output (interleaved re,im) ----
  if (active) {
    float4* o4 = (float4*)(out + (size_t)row * NF * 2 + lane * 64);
#pragma unroll
    for (int k = 0; k < 16; ++k)
      o4[k] = make_float4(s[2 * k].x, s[2 * k].y, s[2 * k + 1].x, s[2 * k + 1].y);
  }
}

// ---------------------------------------------------------------------------
extern "C" void kernel_launch(void* const* d_in, const int* in_sizes, int n_in,
                              void* d_out, int out_size, void* d_ws,
                              size_t ws_size, hipStream_t stream) {
  const float* x      = (const float*)d_in[0];  // (2048,1024) f32
  const float* theta  = (const float*)d_in[1];  // (1,1024,512) f32
  const float* split  = (const float*)d_in[2];  // (1024,512) f32
  const float* atten  = (const float*)d_in[3];  // (1024,1024) c64 -> interleaved f32
  // d_in[4] = index: fixed alternating A/B pattern; derived from layer parity.
  float* out = (float*)d_out;                   // (2048,1024) c64 interleaved

  const int rows = in_sizes[0] / NF;            // 2048
  const int grid = (rows + WAVES_PER_WG - 1) / WAVES_PER_WG;

  const size_t tab_bytes = (size_t)NL * NA * 4 * sizeof(float);  // 8 MB
  if (ws_size >= tab_bytes) {
    float* tab = (float*)d_ws;
    trig_prep<<<(NL * NA + 255) / 256, 256, 0, stream>>>(theta, split, tab, NL * NA);
    clements_scan<true><<<grid, WG_THREADS, 0, stream>>>(x, theta, split, atten,
                                                         tab, out, rows);
  } else {
    clements_scan<false><<<grid, WG_THREADS, 0, stream>>>(x, theta, split, atten,
                                                          nullptr, out, rows);
  }
}